// RSR_67585605370199
// MI455X (gfx1250) — compile-verified
//
#include <hip/hip_runtime.h>
#include <cstddef>

typedef float v2f __attribute__((ext_vector_type(2)));
typedef float v8f __attribute__((ext_vector_type(8)));

#define NROW 1024
#define DF 6
#define TT 60
#define HID 64
#define RR 60
#define G3 192            // 3*HID
#define GRU_THREADS 384   // 12 waves: one per 16-col gate tile
#define X_STRIDE 360      // DF*TT
#define H_STRIDE 68       // 64 + pad (stride%64==4 -> conflict-free wmma A reads)
#define G_STRIDE 196      // 192 + pad
#define GIN_STRIDE 68

#define AROWS 8           // attention rows per block (128 blocks -> more WGPs streaming)
#define P_STRIDE 1028     // 1024 + pad (stride%64==4 -> conflict-free wmma A reads)

// ---------------------------------------------------------------------------
// GRU: each block owns 16 rows, runs all 60 timesteps locally.
// gh = h @ W_hh.T via V_WMMA_F32_16X16X4_F32 (fp32, K=64 -> 16 k-steps).
// ---------------------------------------------------------------------------
__global__ __launch_bounds__(GRU_THREADS) void gru_kernel(
    const float* __restrict__ x,
    const float* __restrict__ W_ih,
    const float* __restrict__ W_hh,
    const float* __restrict__ b_ih,
    const float* __restrict__ b_hh,
    const float* __restrict__ w_att,
    float* __restrict__ h_last,
    float* __restrict__ s_i,
    float* __restrict__ s_j)
{
  __shared__ float sWih[G3 * DF];          //  4608 B
  __shared__ float sBih[G3];               //   768 B
  __shared__ float sBhh[G3];               //   768 B
  __shared__ float sX[16 * X_STRIDE];      // 23040 B
  __shared__ float sH[16 * H_STRIDE];      //  4352 B
  __shared__ float sG[16 * G_STRIDE];      // 12544 B
  __shared__ float sGin[16 * GIN_STRIDE];  //  4352 B   (~50 KB total)

  const int tid  = threadIdx.x;
  const int wave = tid >> 5;
  const int lane = tid & 31;
  const int half = lane >> 4;   // wmma K-half select
  const int l16  = lane & 15;
  const int row0 = blockIdx.x * 16;

  for (int i = tid; i < G3 * DF; i += GRU_THREADS) sWih[i] = W_ih[i];
  for (int i = tid; i < G3; i += GRU_THREADS) { sBih[i] = b_ih[i]; sBhh[i] = b_hh[i]; }
  for (int i = tid; i < 16 * X_STRIDE; i += GRU_THREADS)
    sX[i] = x[(size_t)row0 * X_STRIDE + i];
  for (int i = tid; i < 16 * H_STRIDE; i += GRU_THREADS) sH[i] = 0.f;
  __syncthreads();

  const int colb  = wave * 16;        // gate-column tile base (0..176)
  const int mycol = colb + l16;       // this lane's output column

  for (int t = 0; t < TT; ++t) {
    v8f c;
    // gi = xt @ W_ih.T + b_ih, folded into C init for r/z gates; the n-gate
    // keeps i_n separate (reference: tanh(i_n + r * h_n), h_n includes b_hh).
    #pragma unroll
    for (int v = 0; v < 8; ++v) {
      const int rv = v + 8 * half;                    // C-layout row
      float gi = sBih[mycol];
      #pragma unroll
      for (int f = 0; f < DF; ++f)
        gi += sX[rv * X_STRIDE + f * TT + t] * sWih[mycol * DF + f];
      if (wave < 8) {
        c[v] = gi + sBhh[mycol];                      // r/z: i-part + both biases
      } else {
        sGin[rv * GIN_STRIDE + (mycol - 2 * HID)] = gi; // n-gate i-part
        c[v] = sBhh[mycol];                           // accumulator = gh_n + b_hh
      }
    }
    // gh accumulation: A = h tile (16x64), B = W_hh.T (64x16 slice), K in 4s.
    #pragma unroll
    for (int ks = 0; ks < 16; ++ks) {
      const int k0 = ks * 4 + half * 2;
      v2f a, b;
      a.x = sH[l16 * H_STRIDE + k0];
      a.y = sH[l16 * H_STRIDE + k0 + 1];
      b.x = W_hh[mycol * HID + k0];       // B[k][col] = W_hh[col][k]  (loop-invariant)
      b.y = W_hh[mycol * HID + k0 + 1];
      c = __builtin_amdgcn_wmma_f32_16x16x4_f32(false, a, false, b,
                                                (short)0, c, false, false);
    }
    #pragma unroll
    for (int v = 0; v < 8; ++v)
      sG[(v + 8 * half) * G_STRIDE + mycol] = c[v];
    __syncthreads();

    // elementwise gates: 16*64 states over 384 threads
    for (int e = tid; e < 16 * HID; e += GRU_THREADS) {
      const int n = e >> 6, j = e & 63;
      const float rg = sG[n * G_STRIDE + j];
      const float zg = sG[n * G_STRIDE + HID + j];
      const float hn = sG[n * G_STRIDE + 2 * HID + j];   // gh_n + b_hh
      const float r  = 1.f / (1.f + __expf(-rg));
      const float z  = 1.f / (1.f + __expf(-zg));
      const float nn = tanhf(sGin[n * GIN_STRIDE + j] + r * hn);
      sH[n * H_STRIDE + j] = (1.f - z) * nn + z * sH[n * H_STRIDE + j];
    }
    __syncthreads();
  }

  for (int e = tid; e < 16 * HID; e += GRU_THREADS) {
    const int n = e >> 6, j = e & 63;
    h_last[(size_t)(row0 + n) * HID + j] = sH[n * H_STRIDE + j];
  }
  // s_i / s_j for the block's 16 rows: lanes 0-15 -> s_i, 16-31 -> s_j
  if (wave == 0) {
    float acc = 0.f;
    const int off = half * HID;
    for (int k = 0; k < HID; ++k) acc += sH[l16 * H_STRIDE + k] * w_att[off + k];
    if (half == 0) s_i[row0 + l16] = acc;
    else           s_j[row0 + l16] = acc;
  }
}

// ---------------------------------------------------------------------------
// Attention: one streaming pass over the 252 MB relation tensor per 8-row
// tile (s_rel + mask fused), leaky-relu / sentinel / softmax in LDS, then
// agg = P @ h_last via WMMA (rows 8-15 of the A tile fed as zeros), and the
// final FC folded in via cross-lane reductions. 128 blocks keep more WGPs
// streaming at the HBM roofline.
// ---------------------------------------------------------------------------
__global__ __launch_bounds__(512) void attn_kernel(
    const float* __restrict__ rel,
    const float* __restrict__ w_att,
    const float* __restrict__ b_att,
    const float* __restrict__ W_fc,
    const float* __restrict__ b_fc,
    const float* __restrict__ h_last,
    const float* __restrict__ s_i,
    const float* __restrict__ s_j,
    float* __restrict__ out)
{
  __shared__ float sP[AROWS * P_STRIDE];   // ~33 KB, padded stride
  const int tid  = threadIdx.x;
  const int wave = tid >> 5;
  const int lane = tid & 31;
  const int half = lane >> 4;
  const int l16  = lane & 15;
  const int i0   = blockIdx.x * AROWS;

  const float  batt = b_att[0];
  const float4* wr4 = (const float4*)(w_att + 2 * HID);  // w_rel, uniform -> s_load

  // Phase A: stream relation matrix; compute s_rel dot + mask sum per (i,j)
  for (int p = tid; p < AROWS * 1024; p += 512) {
    const int nl = p >> 10, j = p & 1023;
    const float4* rp = (const float4*)(rel + ((size_t)(i0 + nl) * 1024 + (size_t)j) * RR);
    const int pn = p + 512;
    if (pn < AROWS * 1024) {
      const int nl2 = pn >> 10, j2 = pn & 1023;
      __builtin_prefetch(rel + ((size_t)(i0 + nl2) * 1024 + (size_t)j2) * RR, 0, 1);
    }
    float dot = 0.f, msum = 0.f;
    #pragma unroll
    for (int q = 0; q < 15; ++q) {
      const float4 v  = rp[q];
      const float4 wv = wr4[q];
      dot  += v.x * wv.x + v.y * wv.y + v.z * wv.z + v.w * wv.w;
      msum += v.x + v.y + v.z + v.w;
    }
    float w = s_i[i0 + nl] + s_j[j] + dot + batt;
    w = (w >= 0.f) ? w : 0.01f * w;                 // leaky relu
    const float temp = (msum != 0.f) ? w : 0.f;     // mask * weight
    sP[nl * P_STRIDE + j] = (temp == 0.f) ? -10000.f : temp;
  }
  __syncthreads();

  // Phase B: per-row softmax (wave r owns row r, r < AROWS), masked -> 0
  if (wave < AROWS) {
    const int r = wave;
    float m = -3.4e38f;
    for (int j = lane; j < 1024; j += 32) m = fmaxf(m, sP[r * P_STRIDE + j]);
    #pragma unroll
    for (int s2 = 16; s2 > 0; s2 >>= 1) m = fmaxf(m, __shfl_xor(m, s2, 32));
    float sum = 0.f;
    for (int j = lane; j < 1024; j += 32) sum += __expf(sP[r * P_STRIDE + j] - m);
    #pragma unroll
    for (int s2 = 16; s2 > 0; s2 >>= 1) sum += __shfl_xor(sum, s2, 32);
    const float inv = 1.f / sum;
    for (int j = lane; j < 1024; j += 32) {
      const int idx = r * P_STRIDE + j;
      const float tv = sP[idx];
      sP[idx] = (tv == -10000.f) ? 0.f : __expf(tv - m) * inv;  // softmax * mask
    }
  }
  __syncthreads();

  // Phase C: agg = P(8x1024, zero-padded to 16) @ h_last(1024x64);
  // waves 0-3 own 16 hidden cols each.
  v8f c = {0.f, 0.f, 0.f, 0.f, 0.f, 0.f, 0.f, 0.f};
  if (wave < 4) {
    const int colb = wave * 16;
    const int arow = l16 & 7;          // clamped address; rows 8-15 masked to 0
    const bool live = (l16 < AROWS);
    for (int ks = 0; ks < 256; ++ks) {
      const int k0 = ks * 4 + half * 2;
      v2f a, b;
      const float a0 = sP[arow * P_STRIDE + k0];
      const float a1 = sP[arow * P_STRIDE + k0 + 1];
      a.x = live ? a0 : 0.f;
      a.y = live ? a1 : 0.f;
      b.x = h_last[(size_t)k0 * HID + colb + l16];
      b.y = h_last[(size_t)(k0 + 1) * HID + colb + l16];
      c = __builtin_amdgcn_wmma_f32_16x16x4_f32(false, a, false, b,
                                                (short)0, c, false, false);
    }
  }
  __syncthreads();   // all sP reads done; safe to reuse as scratch

  // Phase D: out[n] = h_last[n].W_fc[:64] + agg[n].W_fc[64:] + b_fc
  float* sOut = sP;  // 64-float scratch aliasing freed P buffer
  if (wave < 4) {
    const float wfc = W_fc[HID + wave * 16 + l16];
    #pragma unroll
    for (int v = 0; v < 8; ++v) {
      float pv = c[v] * wfc;
      pv += __shfl_xor(pv, 1, 32);
      pv += __shfl_xor(pv, 2, 32);
      pv += __shfl_xor(pv, 4, 32);
      pv += __shfl_xor(pv, 8, 32);   // sum within each 16-lane half
      if (l16 == 0) sOut[wave * 16 + v + 8 * half] = pv;  // rows 8-15 are 0
    }
  }
  __syncthreads();
  if (tid < AROWS) {
    const int n = i0 + tid;
    float acc = b_fc[0] + sOut[tid] + sOut[16 + tid] + sOut[32 + tid] + sOut[48 + tid];
    for (int k = 0; k < HID; ++k)
      acc += h_last[(size_t)n * HID + k] * W_fc[k];
    out[n] = acc;
  }
}

// ---------------------------------------------------------------------------
extern "C" void kernel_launch(void* const* d_in, const int* in_sizes, int n_in,
                              void* d_out, int out_size, void* d_ws, size_t ws_size,
                              hipStream_t stream) {
  (void)in_sizes; (void)n_in; (void)out_size; (void)ws_size;
  const float* x     = (const float*)d_in[0];
  const float* rel   = (const float*)d_in[1];
  const float* W_ih  = (const float*)d_in[2];
  const float* W_hh  = (const float*)d_in[3];
  const float* b_ih  = (const float*)d_in[4];
  const float* b_hh  = (const float*)d_in[5];
  const float* w_att = (const float*)d_in[6];
  const float* b_att = (const float*)d_in[7];
  const float* W_fc  = (const float*)d_in[8];
  const float* b_fc  = (const float*)d_in[9];
  float* out = (float*)d_out;

  float* h_last = (float*)d_ws;            // 1024*64 f32
  float* s_i    = h_last + NROW * HID;     // 1024 f32
  float* s_j    = s_i + NROW;              // 1024 f32

  gru_kernel<<<NROW / 16, GRU_THREADS, 0, stream>>>(x, W_ih, W_hh, b_ih, b_hh,
                                                    w_att, h_last, s_i, s_j);
  attn_kernel<<<NROW / AROWS, 512, 0, stream>>>(rel, w_att, b_att, W_fc, b_fc,
                                                h_last, s_i, s_j, out);
}